// ConvCaps2D_23519240912903
// MI455X (gfx1250) — compile-verified
//
#include <hip/hip_runtime.h>
#include <hip/hip_bf16.h>
#include <math.h>

typedef __attribute__((ext_vector_type(2))) float v2f;
typedef __attribute__((ext_vector_type(8))) float v8f;
typedef unsigned int v4u __attribute__((ext_vector_type(4)));
typedef int v4i __attribute__((ext_vector_type(4)));
typedef int v8i __attribute__((ext_vector_type(8)));

#define BATCH 8
#define HH    32
#define WW    32
#define CIN   16
#define OCH   16
#define KH    3
#define KW    3
#define OH    15
#define OW    15
#define NVOTE 144          /* KH*KW*CIN */
#define EL    16           /* 4x4 pose elements */
#define NPIX  (BATCH*OH*OW)

/* LDS layout (float offsets) */
#define OFF_P  0                          /* pose blocks [NVOTE][16]      2304 */
#define OFF_V  (OFF_P  + NVOTE*EL)        /* votes [OCH][NVOTE][16]      36864 */
#define OFF_B  (OFF_V  + OCH*NVOTE*EL)    /* routing logits [OCH][NVOTE]  2304 */
#define OFF_C  (OFF_B  + OCH*NVOTE)       /* coupling coeffs              2304 */
#define OFF_S  (OFF_C  + OCH*NVOTE)       /* s [OCH][16]                   256 */
#define OFF_VC (OFF_S  + OCH*EL)          /* v [OCH][16]                   256 */
#define OFF_SC (OFF_VC + OCH*EL)          /* squash scale [OCH]             16 */
#define LDS_FLOATS (OFF_SC + OCH)         /* = 44304 floats = 177216 B */

extern "C" __global__ __launch_bounds__(256)
void convcaps2d_wmma_kernel(const float* __restrict__ poses,
                            const float* __restrict__ kern,
                            float* __restrict__ out)
{
    extern __shared__ float sm[];
    const int tid  = threadIdx.x;
    const int lane = tid & 31;
    const int wave = tid >> 5;

    const int pix = blockIdx.x;
    const int bb  = pix / (OH * OW);
    const int rem = pix - bb * (OH * OW);
    const int oy  = rem / OW;
    const int ox  = rem - oy * OW;

    /* ---- Phase 0: TDM-stage the receptive field into LDS ----
       Tile: 3 rows (ki, global stride W*CIN*16 = 8192 floats) of 768
       contiguous floats (kj,c,e) each.  Row-major tile order == sP layout. */
    if (wave == 0) {
        const int hi0 = oy * 2;
        const int wi0 = ox * 2;
        const unsigned long long gaddr =
            (unsigned long long)poses +
            (unsigned long long)(((bb * HH + hi0) * WW + wi0) * CIN * EL) * 4ull;

        v4u g0;
        g0[0] = 1u;                                   /* count = 1 valid D# */
        g0[1] = (unsigned)(OFF_P * 4);                /* lds_addr (bytes)   */
        g0[2] = (unsigned)(gaddr & 0xFFFFFFFFu);      /* global_addr[31:0]  */
        g0[3] = (unsigned)((gaddr >> 32) & 0x1FFFFFFu) | (2u << 30); /* type=2 */

        const unsigned td0     = (unsigned)(WW * CIN * EL);  /* tensor_dim0 */
        const unsigned td1     = (unsigned)(BATCH * HH);     /* tensor_dim1 */
        const unsigned tile0   = (unsigned)(KW * CIN * EL);  /* 768 elements */
        const unsigned tile1   = (unsigned)KH;               /* 3 rows       */
        const unsigned stride0 = (unsigned)(WW * CIN * EL);  /* 8192 elems   */

        v8i g1;
        g1[0] = (int)(2u << 16);                      /* mask=0, data_size=4B */
        g1[1] = (int)((td0 & 0xFFFFu) << 16);         /* dim0[15:0] in hi16   */
        g1[2] = (int)(((td0 >> 16) & 0xFFFFu) | ((td1 & 0xFFFFu) << 16));
        g1[3] = (int)(((td1 >> 16) & 0xFFFFu) | (tile0 << 16));  /* tile_dim0 */
        g1[4] = (int)tile1;                           /* tile_dim1, tile_dim2=0 */
        g1[5] = (int)stride0;                         /* dim0_stride[31:0]    */
        g1[6] = 0;
        g1[7] = 0;

        v4i gz = {0, 0, 0, 0};
#if __has_include(<hip/amd_detail/amd_gfx1250_TDM.h>)
        v8i gz8 = {0, 0, 0, 0, 0, 0, 0, 0};
        __builtin_amdgcn_tensor_load_to_lds(g0, g1, gz, gz, gz8, 0);
#else
        __builtin_amdgcn_tensor_load_to_lds(g0, g1, gz, gz, 0);
#endif
        __builtin_amdgcn_s_wait_tensorcnt(0);
    }
    for (int idx = tid; idx < OCH * NVOTE; idx += 256)
        sm[OFF_B + idx] = 0.0f;
    __syncthreads();

    /* ---- Phase 1: votes via V_WMMA_F32_16X16X4_F32 ----
       Wave w owns n in [w*18, w*18+18).  Per n:
         - A(16x4) rows 0..3 = P_n, built once,
         - preload all 4 o-group B operands (8 loads clause together),
         - 4 independent WMMAs back-to-back (distinct accumulators),
         - single lane<16 store region (one exec toggle).                  */
    const int m   = lane & 15;     /* A row / C column / B column */
    const int kh2 = lane >> 4;     /* which K half this lane holds */
    const int r    = m & 3;
    const int osub = m >> 2;
    const int nbase = wave * (NVOTE / 8);
#pragma unroll 2
    for (int dn = 0; dn < NVOTE / 8; ++dn) {
        const int n = nbase + dn;
        /* unconditional LDS load from a safe row, then mask rows 4..15 */
        const float ax = sm[OFF_P + n * 16 + (m & 3) * 4 + 2 * kh2];
        const float ay = sm[OFF_P + n * 16 + (m & 3) * 4 + 2 * kh2 + 1];
        v2f a;
        a.x = (m < 4) ? ax : 0.0f;
        a.y = (m < 4) ? ay : 0.0f;

        v2f bv[4];
#pragma unroll
        for (int og = 0; og < 4; ++og) {
            const float* wp = kern + ((og * 4 + osub) * NVOTE + n) * 16 + r;
            bv[og].x = wp[(2 * kh2)     * 4];
            bv[og].y = wp[(2 * kh2 + 1) * 4];
        }

        v8f cz = {};
        v8f d0 = __builtin_amdgcn_wmma_f32_16x16x4_f32(
            false, a, false, bv[0], (short)0, cz, false, false);
        v8f d1 = __builtin_amdgcn_wmma_f32_16x16x4_f32(
            false, a, false, bv[1], (short)0, cz, false, false);
        v8f d2 = __builtin_amdgcn_wmma_f32_16x16x4_f32(
            false, a, false, bv[2], (short)0, cz, false, false);
        v8f d3 = __builtin_amdgcn_wmma_f32_16x16x4_f32(
            false, a, false, bv[3], (short)0, cz, false, false);

        if (lane < 16) {   /* lanes 0..15 hold C rows 0..7 -> rows 0..3 */
            float* vp0 = &sm[OFF_V + ((0 * 4 + osub) * NVOTE + n) * 16 + r];
            float* vp1 = &sm[OFF_V + ((1 * 4 + osub) * NVOTE + n) * 16 + r];
            float* vp2 = &sm[OFF_V + ((2 * 4 + osub) * NVOTE + n) * 16 + r];
            float* vp3 = &sm[OFF_V + ((3 * 4 + osub) * NVOTE + n) * 16 + r];
            vp0[0] = d0[0]; vp0[4] = d0[1]; vp0[8] = d0[2]; vp0[12] = d0[3];
            vp1[0] = d1[0]; vp1[4] = d1[1]; vp1[8] = d1[2]; vp1[12] = d1[3];
            vp2[0] = d2[0]; vp2[4] = d2[1]; vp2[8] = d2[2]; vp2[12] = d2[3];
            vp3[0] = d3[0]; vp3[4] = d3[1]; vp3[8] = d3[2]; vp3[12] = d3[3];
        }
    }
    __syncthreads();

    /* ---- Phase 2: 3 iterations of dynamic routing (VALU + LDS) ---- */
    for (int it = 0; it < 3; ++it) {
        /* (a) softmax of logits over output capsules, per vote n */
        if (tid < NVOTE) {
            const int n = tid;
            float mx = -1e30f;
            for (int o = 0; o < OCH; ++o)
                mx = fmaxf(mx, sm[OFF_B + o * NVOTE + n]);
            float ex[OCH];
            float ssum = 0.0f;
            for (int o = 0; o < OCH; ++o) {
                ex[o] = __expf(sm[OFF_B + o * NVOTE + n] - mx);
                ssum += ex[o];
            }
            const float inv = 1.0f / ssum;
            for (int o = 0; o < OCH; ++o)
                sm[OFF_C + o * NVOTE + n] = ex[o] * inv;
        }
        __syncthreads();

        /* (b) weighted vote sum: s[o][e] = sum_n c[o][n] * votes[o][n][e] */
        {
            const int o = tid >> 4;
            const int e = tid & 15;
            float acc = 0.0f;
            for (int n = 0; n < NVOTE; ++n)
                acc += sm[OFF_C + o * NVOTE + n] *
                       sm[OFF_V + (o * NVOTE + n) * 16 + e];
            sm[OFF_S + tid] = acc;
        }
        __syncthreads();

        /* (c) squash scale per output capsule */
        if (tid < OCH) {
            float sq = 0.0f;
            for (int e = 0; e < EL; ++e) {
                float x = sm[OFF_S + tid * 16 + e];
                sq += x * x;
            }
            sm[OFF_SC + tid] = (sq / (1.0f + sq)) * __frsqrt_rn(sq + 1e-9f);
        }
        __syncthreads();

        /* (d) v = scale * s */
        sm[OFF_VC + tid] = sm[OFF_S + tid] * sm[OFF_SC + (tid >> 4)];
        __syncthreads();

        /* (e) agreement update: b[o][n] += <v[o], votes[o][n]> */
        if (it < 2) {
            for (int idx = tid; idx < OCH * NVOTE; idx += 256) {
                const int o = idx / NVOTE;
                const float* vv = &sm[OFF_VC + o * 16];
                const float* vt = &sm[OFF_V + idx * 16];
                float acc = 0.0f;
                for (int e = 0; e < EL; ++e)
                    acc += vv[e] * vt[e];
                sm[OFF_B + idx] += acc;
            }
            __syncthreads();
        }
    }

    /* ---- Output: poses [NPIX][16][4][4] then activations [NPIX][16] ---- */
    out[pix * 256 + tid] = sm[OFF_VC + tid];
    if (tid < OCH) {
        float sq = 1e-12f;
        for (int e = 0; e < EL; ++e) {
            float x = sm[OFF_VC + tid * 16 + e];
            sq += x * x;
        }
        out[NPIX * 256 + pix * 16 + tid] = sqrtf(sq);
    }
}

extern "C" void kernel_launch(void* const* d_in, const int* in_sizes, int n_in,
                              void* d_out, int out_size, void* d_ws, size_t ws_size,
                              hipStream_t stream)
{
    (void)in_sizes; (void)n_in; (void)out_size; (void)d_ws; (void)ws_size;

    const float* poses = (const float*)d_in[0];
    /* d_in[1] = input_activations: unused by the reference computation */
    const float* kern  = (const float*)d_in[2];
    float* out = (float*)d_out;

    const size_t smem = (size_t)LDS_FLOATS * sizeof(float);   /* 177216 B */
    (void)hipFuncSetAttribute((const void*)convcaps2d_wmma_kernel,
                              hipFuncAttributeMaxDynamicSharedMemorySize,
                              (int)smem);

    convcaps2d_wmma_kernel<<<NPIX, 256, smem, stream>>>(poses, kern, out);
}